// MyDiag_Model_16003048145332
// MI455X (gfx1250) — compile-verified
//
#include <hip/hip_runtime.h>
#include <hip/hip_bf16.h>
#include <math.h>

// ---------------------------------------------------------------------------
// Problem constants (from reference)
// ---------------------------------------------------------------------------
#define BATCH   32
#define CIN     64
#define H_IN    112
#define W_IN    112
#define HW_IN   (H_IN * W_IN)        // 12544
#define HW_IN4  (HW_IN / 4)          // 3136 float4s per channel plane
#define COUT    128
#define H_OUT   56
#define W_OUT   56
#define HW_OUT  (H_OUT * W_OUT)      // 3136
#define NNODES  16
#define NEDGES  64

typedef float v2f __attribute__((ext_vector_type(2)));
typedef float v4f __attribute__((ext_vector_type(4)));   // native vector: OK for NT builtins
typedef float v8f __attribute__((ext_vector_type(8)));

// D(16x16,f32) = A(16x4,f32) * B(4x16,f32) + C  -- CDNA5 V_WMMA_F32_16X16X4_F32
static __device__ __forceinline__ v8f wmma_f32_16x16x4(v2f a, v2f b, v8f c) {
    return __builtin_amdgcn_wmma_f32_16x16x4_f32(
        /*neg_a=*/false, a, /*neg_b=*/false, b,
        /*c_mod=*/(short)0, c, /*reuse_a=*/false, /*reuse_b=*/false);
}

// ---------------------------------------------------------------------------
// Kernel 0: zero the per-(batch,node) statistics region of the workspace
// ---------------------------------------------------------------------------
__global__ void k_zero(float* p, int n) {
    int i = blockIdx.x * blockDim.x + threadIdx.x;
    if (i < n) p[i] = 0.0f;
}

// ---------------------------------------------------------------------------
// Kernel 1: fused 1x1 conv (Cin=64 -> 1) + superpixel segment accumulation.
// Each thread handles one float4 = 4 consecutive pixels; B_for streamed with
// non-temporal (TH=NT) b128 loads since every byte is touched exactly once.
// sums layout: [b][node][4] = {count, sum_row, sum_col, sum_intensity}
// Grid: (14, 32) x 224 threads: 14*224 = 3136 = HW/4 exactly.
// ---------------------------------------------------------------------------
__global__ void k_conv_seg(const v4f*  __restrict__ Bf4,
                           const int4* __restrict__ seg4,
                           const float* __restrict__ conv_w,
                           const float* __restrict__ conv_b,
                           float*       __restrict__ sums) {
    __shared__ float s_w[CIN];
    __shared__ float s_acc[NNODES * 4];

    const int tid  = threadIdx.x;
    const int b    = blockIdx.y;
    const int pix4 = blockIdx.x * 224 + tid;   // always < 3136

    if (tid < CIN)        s_w[tid]   = conv_w[tid];
    if (tid < NNODES * 4) s_acc[tid] = 0.0f;
    __syncthreads();

    const float cb = conv_b[0];
    const v4f* base = Bf4 + (size_t)b * CIN * HW_IN4 + pix4;
    float L0 = cb, L1 = cb, L2 = cb, L3 = cb;
#pragma unroll
    for (int c = 0; c < CIN; ++c) {
        const v4f v = __builtin_nontemporal_load(&base[(size_t)c * HW_IN4]);
        const float w = s_w[c];
        L0 = fmaf(v.x, w, L0);
        L1 = fmaf(v.y, w, L1);
        L2 = fmaf(v.z, w, L2);
        L3 = fmaf(v.w, w, L3);
    }

    const int4 nd = seg4[(size_t)b * HW_IN4 + pix4];
    const int pix0 = pix4 * 4;                 // W_IN % 4 == 0: same row for all 4
    const int row  = pix0 / W_IN;
    const int col0 = pix0 - row * W_IN;

    const int   nds[4] = { nd.x, nd.y, nd.z, nd.w };
    const float Ls[4]  = { L0, L1, L2, L3 };
#pragma unroll
    for (int j = 0; j < 4; ++j) {
        atomicAdd(&s_acc[nds[j] * 4 + 0], 1.0f);
        atomicAdd(&s_acc[nds[j] * 4 + 1], (float)row);
        atomicAdd(&s_acc[nds[j] * 4 + 2], (float)(col0 + j));
        atomicAdd(&s_acc[nds[j] * 4 + 3], Ls[j]);
    }
    __syncthreads();

    if (tid < NNODES * 4)
        atomicAdd(&sums[b * NNODES * 4 + tid], s_acc[tid]);
}

// ---------------------------------------------------------------------------
// Kernel 2: per-batch graph pipeline. One block = one wave32 per batch.
// Uses V_WMMA_F32_16X16X4_F32 for every matmul.
//
// WMMA operand layouts (wave32, per ISA 7.12.2):
//   A (16x4):  lanes 0-15: v.x=A[m][0], v.y=A[m][1]; lanes 16-31: K=2,3
//   B (4x16):  lanes 0-15: v.x=B[0][n], v.y=B[1][n]; lanes 16-31: K=2,3
//   C/D (16x16): vgpr r, lanes 0-15 -> D[r][n]; lanes 16-31 -> D[r+8][n]
// ---------------------------------------------------------------------------
__global__ void
__launch_bounds__(32)
k_graph(const float* __restrict__ sums,
        const int*   __restrict__ srcp,
        const int*   __restrict__ dstp,
        const float* __restrict__ gW1,  const float* __restrict__ gb1,
        const float* __restrict__ gW2,  const float* __restrict__ gb2,
        const float* __restrict__ gfcw, const float* __restrict__ gfcb,
        float*       __restrict__ Gout) {
    __shared__ float sF[NNODES * 4];      // raw features
    __shared__ float sStat[4];            // mu2, mu3, sd2, sd3
    __shared__ float sA[NNODES * NNODES]; // adjacency
    __shared__ float sNrm[NNODES];        // deg^-1/2
    __shared__ float sYw[NNODES * 16];    // nrm*X, zero-padded to 16x16 (B operand)
    __shared__ float sY[NNODES * 4];      // Z1 = nrm*agg1     (16x4, A operand)
    __shared__ float sD[NNODES * 16];     // WMMA D scratch     (16x16)
    __shared__ float sH[NNODES * 32];     // h1                 (16x32)
    __shared__ float sY2[NNODES * 32];    // nrm*h1 / nrm*agg2  (16x32)
    __shared__ float sD2[NNODES * 32];    // agg2               (16x32)
    __shared__ float sH2[NNODES * COUT];  // h2                 (16x128)

    const int b    = blockIdx.x;
    const int lane = threadIdx.x;         // 0..31
    const int half = lane >> 4;           // 0 or 1
    const int l16  = lane & 15;

    // ---- superpixel features -------------------------------------------
    if (lane < NNODES) {
        const float* s = sums + b * NNODES * 4 + lane * 4;
        float cnt = s[0], sr = s[1], sc = s[2], si = s[3];
        float cs  = fmaxf(cnt, 1.0f);
        sF[lane * 4 + 0] = sr / cs;
        sF[lane * 4 + 1] = sc / cs;
        sF[lane * 4 + 2] = si / cs;
        sF[lane * 4 + 3] = cnt;
    }
    // zero-pad the 16x16 B-operand staging buffer (cols 4..15 stay zero)
    for (int i = lane; i < NNODES * 16; i += 32) sYw[i] = 0.0f;
    __syncthreads();
    if (lane == 0) {
        float m2 = 0.f, m3 = 0.f;
        for (int n = 0; n < NNODES; ++n) { m2 += sF[n*4+2]; m3 += sF[n*4+3]; }
        m2 *= (1.0f / NNODES); m3 *= (1.0f / NNODES);
        float v2 = 0.f, v3 = 0.f;
        for (int n = 0; n < NNODES; ++n) {
            float d2 = sF[n*4+2] - m2, d3 = sF[n*4+3] - m3;
            v2 += d2 * d2; v3 += d3 * d3;
        }
        sStat[0] = m2; sStat[1] = m3;
        sStat[2] = sqrtf(v2 / (NNODES - 1));   // ddof=1
        sStat[3] = sqrtf(v3 / (NNODES - 1));
    }
    // ---- adjacency ------------------------------------------------------
    for (int i = lane; i < NNODES * NNODES; i += 32) sA[i] = 0.0f;
    __syncthreads();
    for (int e = lane; e < NEDGES; e += 32) {
        int s = srcp[b * NEDGES + e];
        int d = dstp[b * NEDGES + e];
        sA[s * NNODES + d] = 1.0f;
        sA[d * NNODES + s] = 1.0f;
    }
    __syncthreads();
    // ---- degree norm + normalized X ------------------------------------
    if (lane < NNODES) {
        float deg = 0.f;
        for (int j = 0; j < NNODES; ++j) deg += sA[lane * NNODES + j];
        float nrm = rsqrtf(fmaxf(deg, 1.0f));
        sNrm[lane] = nrm;
        float f2 = (sF[lane*4+2] - sStat[0]) / (sStat[2] + 1.0f);
        float f3 = (sF[lane*4+3] - sStat[1]) / (sStat[3] + 1.0f);
        sYw[lane*16+0] = nrm * sF[lane*4+0];
        sYw[lane*16+1] = nrm * sF[lane*4+1];
        sYw[lane*16+2] = nrm * f2;
        sYw[lane*16+3] = nrm * f3;
    }
    __syncthreads();

    // ---- agg1 = A @ Y  (16x16)(16x4), B operand zero-padded ------------
    {
        v8f acc = {0.f,0.f,0.f,0.f,0.f,0.f,0.f,0.f};
#pragma unroll
        for (int k = 0; k < 4; ++k) {
            v2f a, bb;
            a.x = sA[l16 * NNODES + 4*k + 2*half + 0];
            a.y = sA[l16 * NNODES + 4*k + 2*half + 1];
            bb.x = sYw[(4*k + 2*half + 0) * 16 + l16];
            bb.y = sYw[(4*k + 2*half + 1) * 16 + l16];
            acc = wmma_f32_16x16x4(a, bb, acc);
        }
#pragma unroll
        for (int r = 0; r < 8; ++r)
            sD[(r + 8*half) * 16 + l16] = acc[r];
    }
    __syncthreads();
    // Z1 = nrm * agg1 (16x4)
    if (lane < NNODES) {
        for (int j = 0; j < 4; ++j)
            sY[lane * 4 + j] = sNrm[lane] * sD[lane * 16 + j];
    }
    __syncthreads();

    // ---- h1 = relu(Z1 @ W1 + b1): (16x4)(4x32) -> 16x32 ----------------
#pragma unroll
    for (int c0 = 0; c0 < 32; c0 += 16) {
        v2f a, bb;
        a.x = sY[l16 * 4 + 2*half + 0];
        a.y = sY[l16 * 4 + 2*half + 1];
        bb.x = gW1[(2*half + 0) * 32 + c0 + l16];
        bb.y = gW1[(2*half + 1) * 32 + c0 + l16];
        v8f acc = {0.f,0.f,0.f,0.f,0.f,0.f,0.f,0.f};
        acc = wmma_f32_16x16x4(a, bb, acc);
        float bias = gb1[c0 + l16];
#pragma unroll
        for (int r = 0; r < 8; ++r)
            sH[(r + 8*half) * 32 + c0 + l16] = fmaxf(acc[r] + bias, 0.0f);
    }
    __syncthreads();
    // Y2 = nrm * h1
    for (int i = lane; i < NNODES * 32; i += 32)
        sY2[i] = sNrm[i >> 5] * sH[i];
    __syncthreads();

    // ---- agg2 = A @ Y2: (16x16)(16x32) ---------------------------------
#pragma unroll
    for (int c0 = 0; c0 < 32; c0 += 16) {
        v8f acc = {0.f,0.f,0.f,0.f,0.f,0.f,0.f,0.f};
#pragma unroll
        for (int k = 0; k < 4; ++k) {
            v2f a, bb;
            a.x = sA[l16 * NNODES + 4*k + 2*half + 0];
            a.y = sA[l16 * NNODES + 4*k + 2*half + 1];
            bb.x = sY2[(4*k + 2*half + 0) * 32 + c0 + l16];
            bb.y = sY2[(4*k + 2*half + 1) * 32 + c0 + l16];
            acc = wmma_f32_16x16x4(a, bb, acc);
        }
#pragma unroll
        for (int r = 0; r < 8; ++r)
            sD2[(r + 8*half) * 32 + c0 + l16] = acc[r];
    }
    __syncthreads();
    // Z2 = nrm * agg2 (reuse sY2)
    for (int i = lane; i < NNODES * 32; i += 32)
        sY2[i] = sNrm[i >> 5] * sD2[i];
    __syncthreads();

    // ---- h2 = relu(Z2 @ W2 + b2): (16x32)(32x128) -> 16x128 ------------
    for (int c0 = 0; c0 < COUT; c0 += 16) {
        v8f acc = {0.f,0.f,0.f,0.f,0.f,0.f,0.f,0.f};
#pragma unroll
        for (int k = 0; k < 8; ++k) {
            v2f a, bb;
            a.x = sY2[l16 * 32 + 4*k + 2*half + 0];
            a.y = sY2[l16 * 32 + 4*k + 2*half + 1];
            bb.x = gW2[(4*k + 2*half + 0) * COUT + c0 + l16];
            bb.y = gW2[(4*k + 2*half + 1) * COUT + c0 + l16];
            acc = wmma_f32_16x16x4(a, bb, acc);
        }
        float bias = gb2[c0 + l16];
#pragma unroll
        for (int r = 0; r < 8; ++r)
            sH2[(r + 8*half) * COUT + c0 + l16] = fmaxf(acc[r] + bias, 0.0f);
    }
    __syncthreads();

    // ---- G[b][f] = sum_n h2[n][f] * fc_w[n] + fc_b ---------------------
    const float fb = gfcb[0];
    for (int f = lane; f < COUT; f += 32) {
        float g = fb;
#pragma unroll
        for (int n = 0; n < NNODES; ++n)
            g = fmaf(sH2[n * COUT + f], gfcw[n], g);
        Gout[b * COUT + f] = g;
    }
}

// ---------------------------------------------------------------------------
// Kernel 3: out[b,c,h,w] = G[b,c] * B_fon[b,c,h,w], float4-vectorized.
// Streaming data (read-once / write-once) uses non-temporal hints.
// HW_OUT = 3136 = 784 float4s per (b,c) channel plane.
// ---------------------------------------------------------------------------
__global__ void k_scale(const v4f*   __restrict__ fon4,
                        const float* __restrict__ G,
                        v4f*         __restrict__ out4,
                        int n4) {
    int i = blockIdx.x * blockDim.x + threadIdx.x;
    if (i >= n4) return;
    const int bc = i / (HW_OUT / 4);       // 784 float4s per channel plane
    const float g = G[bc];                 // reused 784x -> keep cacheable
    v4f v = __builtin_nontemporal_load(&fon4[i]);
    v *= g;
    __builtin_nontemporal_store(v, &out4[i]);
}

// ---------------------------------------------------------------------------
// Launcher
// ---------------------------------------------------------------------------
extern "C" void kernel_launch(void* const* d_in, const int* in_sizes, int n_in,
                              void* d_out, int out_size, void* d_ws, size_t ws_size,
                              hipStream_t stream) {
    const float* B_for  = (const float*)d_in[0];
    const float* B_fon  = (const float*)d_in[1];
    const int*   seg    = (const int*)  d_in[2];
    const int*   src    = (const int*)  d_in[3];
    const int*   dst    = (const int*)  d_in[4];
    const float* conv_w = (const float*)d_in[5];
    const float* conv_b = (const float*)d_in[6];
    const float* W1p    = (const float*)d_in[7];
    const float* b1p    = (const float*)d_in[8];
    const float* W2p    = (const float*)d_in[9];
    const float* b2p    = (const float*)d_in[10];
    const float* fcw    = (const float*)d_in[11];
    const float* fcb    = (const float*)d_in[12];

    float* sums = (float*)d_ws;                      // BATCH*NNODES*4 = 2048 floats
    float* G    = sums + BATCH * NNODES * 4;         // BATCH*COUT     = 4096 floats

    // 0) zero segment sums
    k_zero<<<(BATCH * NNODES * 4 + 255) / 256, 256, 0, stream>>>(sums, BATCH * NNODES * 4);

    // 1) fused conv + segment accumulation (float4 path): 14 chunks x 32 batches
    k_conv_seg<<<dim3(HW_IN4 / 224, BATCH), 224, 0, stream>>>(
        (const v4f*)B_for, (const int4*)seg, conv_w, conv_b, sums);

    // 2) graph pipeline, one wave32 per batch (WMMA path)
    k_graph<<<BATCH, 32, 0, stream>>>(sums, src, dst, W1p, b1p, W2p, b2p, fcw, fcb, G);

    // 3) broadcast scale of B_fon (non-temporal streaming)
    const int n4 = BATCH * COUT * HW_OUT / 4;        // 3,211,264
    k_scale<<<(n4 + 255) / 256, 256, 0, stream>>>((const v4f*)B_fon, G, (v4f*)d_out, n4);
}